// bicon_loss_74139725463774
// MI455X (gfx1250) — compile-verified
//
#include <hip/hip_runtime.h>
#include <hip/hip_bf16.h>

#define BB   8
#define HH   352
#define WW   352
#define HWSZ (HH * WW)        // 123904
#define NPIX (BB * HWSZ)      // 991232 = 3872 * 256 exactly
#define TPB  256
#define NBLK (NPIX / TPB)     // 3872
#define NSAL 4
#define NWORDS (NPIX / 32)    // 30976 packed edge-map words

typedef __attribute__((ext_vector_type(2))) float v2f;
typedef __attribute__((ext_vector_type(8))) float v8f;

// ---------------------------------------------------------------------------
// Wave32 sum via V_WMMA_F32_16X16X4_F32:
//   A (16x4 f32): lane m (0..15) holds {K0,K1}, lane m+16 holds {K2,K3}.
//   Put v in A-VGPR0, 0 in A-VGPR1; B (4x16) = ones  ->  D[m,*] = v_m + v_{m+16}.
//   Per-lane sum of the 8 D VGPRs = sum over M=0..7 (lanes 0..15) or M=8..15
//   (lanes 16..31); one shfl_xor(16) completes the 32-lane sum.
// EXEC must be all ones at the call site.
// ---------------------------------------------------------------------------
__device__ __forceinline__ float wave_sum_wmma(float v) {
  v2f a; a.x = v;    a.y = 0.0f;
  v2f b; b.x = 1.0f; b.y = 1.0f;
  v8f c = {0.f, 0.f, 0.f, 0.f, 0.f, 0.f, 0.f, 0.f};
  c = __builtin_amdgcn_wmma_f32_16x16x4_f32(false, a, false, b,
                                            (short)0, c, false, false);
  float s = c[0] + c[1] + c[2] + c[3] + c[4] + c[5] + c[6] + c[7];
  s += __shfl_xor(s, 16, 32);
  return s;  // full 32-lane sum, replicated in every lane
}

// Block sum (8 waves). Result valid at threadIdx.x == 0. Deterministic order.
__device__ __forceinline__ float block_sum(float v, float* lds) {
  float wsum = wave_sum_wmma(v);
  int lane = threadIdx.x & 31, wid = threadIdx.x >> 5;
  if (lane == 0) lds[wid] = wsum;
  __syncthreads();
  float r = 0.0f;
  if (threadIdx.x == 0) {
    #pragma unroll
    for (int i = 0; i < TPB / 32; ++i) r += lds[i];
  }
  __syncthreads();  // allow immediate LDS reuse
  return r;
}

// Fast transcendental helpers: native v_exp_f32 / v_log_f32 / v_rcp_f32.
// ~1 ulp each; invisible under the f32 accumulation noise of a 1e6-term sum.
__device__ __forceinline__ float sigf(float x) {
  return __fdividef(1.0f, 1.0f + __expf(-x));
}
// softplus(x) = log(1+exp(x)), stable form
__device__ __forceinline__ float softplusf(float x) {
  return fmaxf(x, 0.0f) + __logf(1.0f + __expf(-fabsf(x)));
}

// ---------------------------------------------------------------------------
// K1: edge map. Counts edge pixels (0 < sum_ch con < 8) with an integer
// atomic (order-independent -> deterministic) and packs the per-wave
// 32-pixel edge bitmask into the workspace for K2.
// ---------------------------------------------------------------------------
__global__ void k_count(const int* __restrict__ con, int* __restrict__ counter,
                        unsigned int* __restrict__ emask) {
  int idx = blockIdx.x * TPB + threadIdx.x;           // < NPIX exactly
  int b = idx / HWSZ, hw = idx % HWSZ;
  const int* cp = con + (size_t)b * 8 * HWSZ + hw;
  int s = 0;
  #pragma unroll
  for (int ch = 0; ch < 8; ++ch) s += cp[(size_t)ch * HWSZ];
  int isE = (s > 0 && s < 8) ? 1 : 0;
  unsigned long long m = __ballot(isE);               // wave32: low 32 bits
  __shared__ int cnt[TPB / 32];
  int lane = threadIdx.x & 31, wid = threadIdx.x >> 5;
  if (lane == 0) {
    cnt[wid] = __popcll(m);
    emask[(unsigned)idx >> 5] = (unsigned int)m;      // packed edge map
  }
  __syncthreads();
  if (threadIdx.x == 0) {
    int t = 0;
    #pragma unroll
    for (int i = 0; i < TPB / 32; ++i) t += cnt[i];
    atomicAdd(counter, t);
  }
}

// ---------------------------------------------------------------------------
// K2: edge loss, 3 maps fused. edge flag read from packed bitmask (1 bit/pix
// instead of 8 int32 loads). wts = pos ? num_neg/N : 1.1*num_pos/N.
//     ce = max(x,0) - x*t + log1p(exp(-|x|))
// ---------------------------------------------------------------------------
__global__ void k_edge(const float* __restrict__ up_edge,
                       const unsigned int* __restrict__ emask,
                       const int* __restrict__ counter,
                       float* __restrict__ part) {
  int idx  = blockIdx.x * TPB + threadIdx.x;
  int lane = threadIdx.x & 31;
  unsigned int word = emask[(unsigned)idx >> 5];
  float t = (float)((word >> lane) & 1u);

  float np = (float)(*counter);
  float nn = (float)NPIX - np;
  float w  = (t > 0.5f) ? (nn * (1.0f / (float)NPIX))
                        : (1.1f * np * (1.0f / (float)NPIX));

  float acc = 0.0f;
  #pragma unroll
  for (int i = 0; i < 3; ++i) {
    float x  = up_edge[(size_t)i * NPIX + idx];
    float ce = fmaxf(x, 0.0f) - x * t + __logf(1.0f + __expf(-fabsf(x)));
    acc += w * ce;
  }
  __shared__ float lds[TPB / 32];
  float r = block_sum(acc, lds);
  if (threadIdx.x == 0) part[blockIdx.x] = r;
}

// ---------------------------------------------------------------------------
// K3: saliency loss for map m = blockIdx.y (0..3): m<2 -> up_sal, else up_sal_f.
// vote order (con channel k): v0=a7..v7=a8 per reference stack order.
// ---------------------------------------------------------------------------
__global__ void k_sal(const float* __restrict__ up_sal,
                      const float* __restrict__ up_sal_f,
                      const float* __restrict__ target,
                      const int* __restrict__ con,
                      float* __restrict__ part) {
  int m = blockIdx.y;
  const float* sal = (m < 2) ? (up_sal   + (size_t)m       * 8 * NPIX)
                             : (up_sal_f + (size_t)(m - 2) * 8 * NPIX);

  int idx = blockIdx.x * TPB + threadIdx.x;
  int b = idx / HWSZ, hw = idx % HWSZ;
  int h = hw / WW, w = hw % WW;
  const float* base = sal + (size_t)b * 8 * HWSZ;
  const int*   cp   = con + (size_t)b * 8 * HWSZ + hw;

  float x[8], sgm[8];
  #pragma unroll
  for (int ch = 0; ch < 8; ++ch) {
    x[ch]   = base[(size_t)ch * HWSZ + hw];
    sgm[ch] = sigf(x[ch]);
  }
  int cf[8], csum = 0;
  #pragma unroll
  for (int ch = 0; ch < 8; ++ch) { cf[ch] = cp[(size_t)ch * HWSZ]; csum += cf[ch]; }
  float e = (csum > 0 && csum < 8) ? 1.0f : 0.0f;

  auto nbr = [&](int ch, int hh, int ww2) -> float {
    if (hh < 0 || hh >= HH || ww2 < 0 || ww2 >= WW) return 0.0f;
    return sigf(base[(size_t)ch * HWSZ + hh * WW + ww2]);
  };

  float v[8];
  v[0] = sgm[0] * nbr(7, h - 1, w - 1);  // a7
  v[1] = sgm[1] * nbr(6, h - 1, w    );  // a3
  v[2] = sgm[2] * nbr(5, h - 1, w + 1);  // a5
  v[3] = sgm[3] * nbr(4, h,     w - 1);  // a1
  v[4] = sgm[4] * nbr(3, h,     w + 1);  // a2
  v[5] = sgm[5] * nbr(2, h + 1, w - 1);  // a6
  v[6] = sgm[6] * nbr(1, h + 1, w    );  // a4
  v[7] = sgm[7] * nbr(0, h + 1, w + 1);  // a8

  float vs = 0.0f, vmin = v[0];
  #pragma unroll
  for (int k = 0; k < 8; ++k) { vs += v[k]; vmin = fminf(vmin, v[k]); }
  float glo = 0.125f * vs;
  float dec = (e > 0.5f) ? (1.0f - vmin) : glo;  // glo*(1-e) + (1-vmin)*e

  float t  = target[idx];
  float dl = -(t * fmaxf(__logf(dec), -100.0f) +
               (1.0f - t) * fmaxf(__logf(1.0f - dec), -100.0f));

  float vb = 0.0f, pb = 0.0f;
  #pragma unroll
  for (int k = 0; k < 8; ++k) {
    if (cf[k]) {
      vb += -fmaxf(__logf(v[k]), -100.0f);        // -log(vote)
      pb +=  fminf(softplusf(-x[k]), 100.0f);     // -log(sigmoid(x))
    } else {
      vb += -fmaxf(__logf(1.0f - v[k]), -100.0f); // -log1p(-vote)
      pb +=  fminf(softplusf(x[k]), 100.0f);      // -log(1-sigmoid(x))
    }
  }
  float acc = dl + 0.2f * vb + 0.8f * pb;

  __shared__ float lds[TPB / 32];
  float r = block_sum(acc, lds);
  if (threadIdx.x == 0) part[(size_t)m * NBLK + blockIdx.x] = r;
}

// ---------------------------------------------------------------------------
// K4: deterministic final reduce (fixed per-thread stride order + WMMA tree).
// ---------------------------------------------------------------------------
__global__ void k_final(const float* __restrict__ edgePart,
                        const float* __restrict__ salPart,
                        float* __restrict__ out) {
  __shared__ float lds[TPB / 32];
  float ea = 0.0f;
  for (int i = threadIdx.x; i < NBLK; i += TPB) ea += edgePart[i];
  float edge_l = block_sum(ea, lds);  // valid @ tid 0

  float sa = 0.0f;
  for (int i = threadIdx.x; i < NSAL * NBLK; i += TPB) sa += salPart[i];
  float sal_l = block_sum(sa, lds);   // valid @ tid 0

  if (threadIdx.x == 0) {
    float loss = edge_l + sal_l;
    out[0] = loss; out[1] = edge_l; out[2] = sal_l; out[3] = loss;
  }
}

// ---------------------------------------------------------------------------
extern "C" void kernel_launch(void* const* d_in, const int* in_sizes, int n_in,
                              void* d_out, int out_size, void* d_ws, size_t ws_size,
                              hipStream_t stream) {
  (void)in_sizes; (void)n_in; (void)out_size; (void)ws_size;
  const float* up_edge  = (const float*)d_in[0];  // (3,B,1,H,W)
  const float* up_sal   = (const float*)d_in[1];  // (2,B,8,H,W)
  const float* up_sal_f = (const float*)d_in[2];  // (2,B,8,H,W)
  const float* target   = (const float*)d_in[3];  // (B,1,H,W)
  const int*   con      = (const int*)d_in[4];    // (B,8,H,W)

  int*          counter  = (int*)d_ws;                              // 4 B
  float*        edgePart = (float*)((char*)d_ws + 256);             // NBLK f32
  float*        salPart  = edgePart + NBLK;                         // 4*NBLK f32
  unsigned int* emask    = (unsigned int*)(salPart + NSAL * NBLK);  // NWORDS u32
  float*        out      = (float*)d_out;

  hipMemsetAsync(d_ws, 0, 256, stream);
  k_count<<<NBLK, TPB, 0, stream>>>(con, counter, emask);
  k_edge <<<NBLK, TPB, 0, stream>>>(up_edge, emask, counter, edgePart);
  dim3 gs(NBLK, NSAL);
  k_sal  <<<gs,   TPB, 0, stream>>>(up_sal, up_sal_f, target, con, salPart);
  k_final<<<1,    TPB, 0, stream>>>(edgePart, salPart, out);
}